// NewsvendorLayer_61400852464227
// MI455X (gfx1250) — compile-verified
//
#include <hip/hip_runtime.h>
#include <stdint.h>

typedef __attribute__((ext_vector_type(2))) float v2f;
typedef __attribute__((ext_vector_type(8))) float v8f;

#define TAU_C   0.6f
#define GAMMA_C 0.1f
#define NITER   60
#define EPS_C   1e-12f
#define SMAX    1024      // padded S (actual S = 1001)
#define TPB     128       // 4 waves

static __device__ __forceinline__ int imin(int a, int b) { return a < b ? a : b; }

// g(z) and g'(z) from moments (A,Bm,Q) and the CDF value C(z).
static __device__ __forceinline__ void grad_z(float z, float C, float A, float Bm,
                                              float Q, float* g, float* gp) {
  float s1  = A - z * Bm;
  float q   = fmaf(z, fmaf(z, Bm, -2.0f * A), Q) + EPS_C;  // Q - 2zA + z^2 B + eps
  float nrm = sqrtf(q);
  *g  = (C - TAU_C) - GAMMA_C * s1 / nrm;
  *gp = GAMMA_C * (Bm / nrm - (s1 * s1) / (nrm * nrm * nrm));
}

__launch_bounds__(TPB)
__global__ void newsvendor_solve_kernel(const float* __restrict__ x,
                                        const float* __restrict__ ysup,
                                        float* __restrict__ out,
                                        int B, int S) {
  __shared__ float sp[SMAX];     // p row
  __shared__ float sy[SMAX];     // support grid
  __shared__ float sI[SMAX];     // inclusive prefix sums of p
  __shared__ float stot[TPB];    // per-thread scan totals
  __shared__ float smom[12];     // 4 waves x 3 moments

  const int tid  = threadIdx.x;
  const int row  = blockIdx.x;
  const int lane = tid & 31;
  const int wave = tid >> 5;

  // ---- Stage row of x and y_supp into LDS via CDNA5 async global->LDS ----
  {
    unsigned long long xbase = (unsigned long long)(x + (size_t)row * S);
    unsigned long long ybase = (unsigned long long)ysup;
    const int nq = S >> 2;                       // full 16-byte chunks (250)
    for (int c = tid; c < nq; c += TPB) {
      uint32_t go   = (uint32_t)(c * 16);
      uint32_t ldsp = (uint32_t)(uintptr_t)&sp[c * 4];  // low 32 bits of flat LDS addr = LDS offset
      uint32_t ldsy = (uint32_t)(uintptr_t)&sy[c * 4];
      asm volatile("global_load_async_to_lds_b128 %0, %1, %2 offset:0"
                   :: "v"(ldsp), "v"(go), "s"(xbase) : "memory");
      asm volatile("global_load_async_to_lds_b128 %0, %1, %2 offset:0"
                   :: "v"(ldsy), "v"(go), "s"(ybase) : "memory");
    }
    for (int i = (nq << 2) + tid; i < S; i += TPB) {    // b32 tail (1 element)
      uint32_t go   = (uint32_t)(i * 4);
      uint32_t ldsp = (uint32_t)(uintptr_t)&sp[i];
      uint32_t ldsy = (uint32_t)(uintptr_t)&sy[i];
      asm volatile("global_load_async_to_lds_b32 %0, %1, %2 offset:0"
                   :: "v"(ldsp), "v"(go), "s"(xbase) : "memory");
      asm volatile("global_load_async_to_lds_b32 %0, %1, %2 offset:0"
                   :: "v"(ldsy), "v"(go), "s"(ybase) : "memory");
    }
    asm volatile("s_wait_asynccnt 0" ::: "memory");
  }
  __syncthreads();
  // zero-pad [S, SMAX) so the WMMA moment loop needs no guards
  for (int i = S + tid; i < SMAX; i += TPB) { sp[i] = 0.0f; sy[i] = 0.0f; }
  __syncthreads();

  // ---- Moments via v_wmma_f32_16x16x4_f32 ----
  // A-tile: rows m=0,1,2 hold (p^2, p^2*y, p^2*y^2) for 4 consecutive elements (K dim).
  // B-tile: all ones => D[m,n] = sum_k A[m,k], accumulated over 64 chunks per wave.
  // Lane weight polynomial (branchless): w(y) = f0 + f1*y + f2*y^2 with 0/1 flags.
  {
    const int m  = lane & 15;
    const int hi = lane >> 4;
    const float f0 = (m == 0) ? 1.0f : 0.0f;
    const float f1 = (m == 1) ? 1.0f : 0.0f;
    const float f2 = (m == 2) ? 1.0f : 0.0f;
    const int qbeg = wave * (SMAX / 4);          // each wave: 256 padded elements
    v8f acc = {};
    v2f bv;  bv.x = 1.0f;  bv.y = 1.0f;
    for (int it = 0; it < (SMAX / 4) / 4; ++it) {
      int i0 = qbeg + it * 4 + 2 * hi;           // A layout: lanes<16 -> K=0,1 ; lanes>=16 -> K=2,3
      int i1 = i0 + 1;
      float p0 = sp[i0], y0 = sy[i0], u0 = p0 * p0;
      float p1 = sp[i1], y1 = sy[i1], u1 = p1 * p1;
      v2f av;
      av.x = u0 * fmaf(y0, fmaf(y0, f2, f1), f0);
      av.y = u1 * fmaf(y1, fmaf(y1, f2, f1), f0);
      acc = __builtin_amdgcn_wmma_f32_16x16x4_f32(false, av, false, bv,
                                                  (short)0, acc, false, false);
    }
    // D layout: VGPR v, lanes 0-15 -> M=v, N=lane. Lane 0 holds D[0..7, 0].
    if (lane == 0) {
      smom[wave * 3 + 0] = acc[0];   // sum p^2         (quarter)
      smom[wave * 3 + 1] = acc[1];   // sum p^2 y
      smom[wave * 3 + 2] = acc[2];   // sum p^2 y^2
    }
  }

  // ---- Inclusive prefix sums of p (block scan: 8 elems/thread) ----
  const int CH  = SMAX / TPB;                    // 8
  const int beg = tid * CH;
  const int end = imin(beg + CH, S);
  {
    float run = 0.0f;
    for (int i = beg; i < end; ++i) { run += sp[i]; sI[i] = run; }
    stot[tid] = run;
  }
  __syncthreads();
  if (tid == 0) {
    float acc = 0.0f;
    for (int t = 0; t < TPB; ++t) { float v = stot[t]; stot[t] = acc; acc += v; }
  }
  __syncthreads();
  {
    float off = stot[tid];
    for (int i = beg; i < end; ++i) sI[i] += off;
  }
  __syncthreads();

  // ---- Bisection + Newton (all threads redundantly; all LDS reads broadcast) ----
  const float Bm = smom[0] + smom[3] + smom[6] + smom[9];
  const float Am = smom[1] + smom[4] + smom[7] + smom[10];
  const float Qm = smom[2] + smom[5] + smom[8] + smom[11];

  // C(z) = inclusive-prefix[count(y_s < z) - 1]
  auto cdf = [&](float z) -> float {
    int l = 0, h = S;
    while (l < h) { int mid = (l + h) >> 1; if (sy[mid] < z) l = mid + 1; else h = mid; }
    return (l == 0) ? 0.0f : sI[l - 1];
  };

  float lo = 0.0f, hi = 1.0f;
  for (int it = 0; it < NITER; ++it) {
    float mid = 0.5f * (lo + hi);
    float g, gp;
    grad_z(mid, cdf(mid), Am, Bm, Qm, &g, &gp);
    if (g < 0.0f) lo = mid; else hi = mid;
  }
  float z0 = 0.5f * (lo + hi);

  float g, gp;
  grad_z(z0, cdf(z0), Am, Bm, Qm, &g, &gp);
  float z = z0 - g / fmaxf(gp, 1e-8f);

  // boundary clamp: g(0) = -tau - gamma*A/sqrt(Q+eps) (C(0)=0); g(1) via cdf
  float g0 = -TAU_C - GAMMA_C * Am / sqrtf(Qm + EPS_C);
  float g1, gp1;
  grad_z(1.0f, cdf(1.0f), Am, Bm, Qm, &g1, &gp1);
  z = (g0 >= 0.0f) ? 0.0f
      : (g1 <= 0.0f) ? 1.0f
                     : fminf(fmaxf(z, 0.0f), 1.0f);

  // ---- Streamed outputs (non-temporal: 394 MB, write-once) ----
  const size_t BS = (size_t)B * (size_t)S;
  float* o_pb = out + B;
  float* o_e  = out + B + BS;
  float* o_w  = out + B + 2 * BS;
  const size_t rb = (size_t)row * (size_t)S;

  if (tid == 0) out[row] = z;
  for (int i = tid; i < S; i += TPB) {
    float yv = sy[i];
    float ev = yv - z;
    float pb = fmaxf(TAU_C * ev, (TAU_C - 1.0f) * ev);
    float wv = sp[i] * ev;
    __builtin_nontemporal_store(pb, o_pb + rb + i);
    __builtin_nontemporal_store(ev, o_e  + rb + i);
    __builtin_nontemporal_store(wv, o_w  + rb + i);
  }
}

extern "C" void kernel_launch(void* const* d_in, const int* in_sizes, int n_in,
                              void* d_out, int out_size, void* d_ws, size_t ws_size,
                              hipStream_t stream) {
  (void)n_in; (void)out_size; (void)d_ws; (void)ws_size;
  const float* x = (const float*)d_in[0];
  const float* y = (const float*)d_in[1];
  float* out = (float*)d_out;
  const int S = in_sizes[1];            // 1001
  const int B = in_sizes[0] / S;        // 32768
  newsvendor_solve_kernel<<<B, TPB, 0, stream>>>(x, y, out, B, S);
}